// GATModel_61804579389668
// MI455X (gfx1250) — compile-verified
//
#include <hip/hip_runtime.h>
#include <hip/hip_bf16.h>

// ---------------------------------------------------------------------------
// GAT (3x GATConv + linear classifier) for MI455X / gfx1250 (wave32, WMMA).
// GEMMs use V_WMMA_F32_16X16X4_F32 (one wave per 16x16 output tile).
// Edge softmax/aggregation are memory-bound scatter kernels with atomics.
// ---------------------------------------------------------------------------

typedef float v2f __attribute__((ext_vector_type(2)));
typedef float v8f __attribute__((ext_vector_type(8)));

#define GAT_N      50000
#define GAT_DIN    128
#define GAT_HID    64
#define GAT_HEADS  8
#define GAT_DOUT   10

// ---------------------------------------------------------------------------
// GEMM: Hout[M, Ncol] = X[M, K] @ W[Ncol, K]^T     (fp32 via WMMA 16x16x4)
// One wave32 computes one 16x16 tile. M, Ncol multiples of 16 guaranteed by
// grid sizing (waves whose tile is out of range exit wave-uniformly, so EXEC
// stays all-ones around the WMMA ops).
// ---------------------------------------------------------------------------
__global__ void __launch_bounds__(256)
gat_wmma_gemm_xwT(const float* __restrict__ X, const float* __restrict__ W,
                  float* __restrict__ Hout, int M, int K, int Ncol) {
    const int gtid    = blockIdx.x * blockDim.x + threadIdx.x;
    const int wave_id = gtid >> 5;
    const int lane    = threadIdx.x & 31;

    const int tilesN = Ncol >> 4;
    const int tM = (wave_id / tilesN) << 4;
    const int tN = (wave_id % tilesN) << 4;
    if (tM >= M) return;                       // wave-uniform exit

    // A-frag layout (32-bit A 16x4): lanes 0-15 -> K = k0+{0,1},
    //                                lanes 16-31 -> K = k0+{2,3}, M = lane&15.
    // B-frag (4x16, B = W^T): same K split, N = lane&15.
    const int mrow = tM + (lane & 15);
    const int ncol = tN + (lane & 15);
    const int koff = (lane >> 4) << 1;         // 0 for lanes 0-15, 2 for 16-31

    const float* __restrict__ xrow = X + (size_t)mrow * K + koff;
    const float* __restrict__ wrow = W + (size_t)ncol * K + koff;

    v8f acc = {};
    for (int k0 = 0; k0 < K; k0 += 4) {
        v2f a = *(const v2f*)(xrow + k0);      // X[mrow][k0+koff .. +1]
        v2f b = *(const v2f*)(wrow + k0);      // W[ncol][k0+koff .. +1]
        acc = __builtin_amdgcn_wmma_f32_16x16x4_f32(
            /*neg_a=*/false, a, /*neg_b=*/false, b,
            /*c_mod=*/(short)0, acc, /*reuse_a=*/false, /*reuse_b=*/false);
    }

    // C/D layout: VGPR r -> row (r + 8*(lane>=16)), col = lane&15.
    const int r0 = tM + ((lane >> 4) << 3);
    float* __restrict__ out = Hout + (size_t)r0 * Ncol + (tN + (lane & 15));
#pragma unroll
    for (int r = 0; r < 8; ++r) out[(size_t)r * Ncol] = acc[r];
}

// ---------------------------------------------------------------------------
// Per-node attention coefficients + init of segment max / sum.
// One thread per (node, head); dot of 64 channels.
// ---------------------------------------------------------------------------
__global__ void gat_attn_coeff(const float* __restrict__ H,
                               const float* __restrict__ a_src,
                               const float* __restrict__ a_dst,
                               float* __restrict__ as, float* __restrict__ ad,
                               float* __restrict__ m,  float* __restrict__ s,
                               int n_nodes, int heads) {
    int i = blockIdx.x * blockDim.x + threadIdx.x;
    if (i >= n_nodes * heads) return;
    int node = i / heads, hh = i - node * heads;
    const float* hp  = H + ((size_t)node * heads + hh) * GAT_HID;
    const float* asr = a_src + hh * GAT_HID;
    const float* adr = a_dst + hh * GAT_HID;
    float vs = 0.f, vd = 0.f;
#pragma unroll 8
    for (int c = 0; c < GAT_HID; ++c) { float x = hp[c]; vs += x * asr[c]; vd += x * adr[c]; }
    as[i] = vs;
    ad[i] = vd;
    m[i]  = __uint_as_float(0xff800000u);   // -inf
    s[i]  = 0.f;
}

__global__ void gat_zero(float* __restrict__ p, size_t n) {
    size_t i = (size_t)blockIdx.x * blockDim.x + threadIdx.x;
    if (i < n) p[i] = 0.f;
}

// ---------------------------------------------------------------------------
// Edge helpers. Edge list = [edge_index, self-loops]: e < E uses edge_index
// rows (src at [0,E), dst at [E,2E)); e >= E is self loop (e-E, e-E).
// ---------------------------------------------------------------------------
__device__ __forceinline__ void edge_sd(const int* __restrict__ ei, int E, int e,
                                        int& sn, int& dn) {
    if (e < E) { sn = ei[e]; dn = ei[E + e]; }
    else       { sn = e - E; dn = e - E; }
}

__device__ __forceinline__ float leaky02(float x) { return x > 0.f ? x : 0.2f * x; }

// float atomic max via sign-split on int/uint atomics (init must be -inf).
__device__ __forceinline__ void atomicMaxF(float* addr, float val) {
    if (val >= 0.f) atomicMax((int*)addr, __float_as_int(val));
    else            atomicMin((unsigned int*)addr, __float_as_uint(val));
}

__global__ void gat_edge_max(const int* __restrict__ ei, int E, int Etot,
                             const float* __restrict__ as, const float* __restrict__ ad,
                             float* __restrict__ m, int heads) {
    int i = blockIdx.x * blockDim.x + threadIdx.x;
    if (i >= Etot * heads) return;
    int e = i / heads, hh = i - e * heads;
    int sn, dn; edge_sd(ei, E, e, sn, dn);
    float v = leaky02(as[sn * heads + hh] + ad[dn * heads + hh]);
    atomicMaxF(&m[dn * heads + hh], v);
}

__global__ void gat_edge_sum(const int* __restrict__ ei, int E, int Etot,
                             const float* __restrict__ as, const float* __restrict__ ad,
                             const float* __restrict__ m, float* __restrict__ s,
                             int heads) {
    int i = blockIdx.x * blockDim.x + threadIdx.x;
    if (i >= Etot * heads) return;
    int e = i / heads, hh = i - e * heads;
    int sn, dn; edge_sd(ei, E, e, sn, dn);
    int idx = dn * heads + hh;
    float v = leaky02(as[sn * heads + hh] + ad[idx]);
    atomicAdd(&s[idx], __expf(v - m[idx]));
}

// One wave32 per (edge, head): lanes cover the 64 channels (2 each).
// Alpha is recomputed per wave (uniform) instead of materializing 27MB of
// per-edge weights -- VALU recompute is free vs. HBM scatter traffic.
__global__ void __launch_bounds__(256)
gat_edge_aggr(const int* __restrict__ ei, int E, int Etot,
              const float* __restrict__ as, const float* __restrict__ ad,
              const float* __restrict__ m,  const float* __restrict__ s,
              const float* __restrict__ H,  float* __restrict__ Out, int heads) {
    int gtid = blockIdx.x * blockDim.x + threadIdx.x;
    int wid  = gtid >> 5;
    int lane = threadIdx.x & 31;
    if (wid >= Etot * heads) return;
    int e = wid / heads, hh = wid - e * heads;
    int sn, dn; edge_sd(ei, E, e, sn, dn);
    int idx = dn * heads + hh;
    float v     = leaky02(as[sn * heads + hh] + ad[idx]);
    float alpha = __expf(v - m[idx]) / (s[idx] + 1e-16f);
    const float* hp = H   + ((size_t)sn * heads + hh) * GAT_HID;
    float*       op = Out + ((size_t)dn * heads + hh) * GAT_HID;
    atomicAdd(&op[lane],      hp[lane]      * alpha);
    atomicAdd(&op[lane + 32], hp[lane + 32] * alpha);
}

__global__ void gat_bias_elu(float* __restrict__ Out, const float* __restrict__ b,
                             int n_nodes, int HC) {
    int i = blockIdx.x * blockDim.x + threadIdx.x;
    if (i >= n_nodes * HC) return;
    int j = i % HC;
    float x = Out[i] + b[j];
    Out[i] = x > 0.f ? x : (__expf(x) - 1.f);
}

__global__ void gat_classifier(const float* __restrict__ H,
                               const float* __restrict__ Wc,
                               const float* __restrict__ bc,
                               float* __restrict__ out, int n_nodes) {
    int i = blockIdx.x * blockDim.x + threadIdx.x;
    if (i >= n_nodes * GAT_DOUT) return;
    int n = i / GAT_DOUT, o = i - n * GAT_DOUT;
    const float* hp = H + (size_t)n * GAT_HID;
    const float* w  = Wc + o * GAT_HID;
    float acc = bc[o];
#pragma unroll 8
    for (int c = 0; c < GAT_HID; ++c) acc += hp[c] * w[c];
    out[i] = acc;
}

// ---------------------------------------------------------------------------
// Host-side orchestration (all on `stream`; graph-capture safe).
// ---------------------------------------------------------------------------
static inline int cdiv(long long a, long long b) { return (int)((a + b - 1) / b); }

static void run_gat_layer(const float* Xin, const float* W, const float* a_src,
                          const float* a_dst, const float* bias,
                          float* bufH, float* bufO,
                          float* as, float* ad, float* mm, float* ss,
                          const int* ei, int E, int Etot,
                          int n_nodes, int K, int heads, hipStream_t stream) {
    const int HC = heads * GAT_HID;
    // 1) h = x @ W^T  via WMMA
    {
        long long waves = (long long)(n_nodes / 16) * (HC / 16);
        gat_wmma_gemm_xwT<<<cdiv(waves, 8), 256, 0, stream>>>(Xin, W, bufH, n_nodes, K, HC);
    }
    // 2) attention coefficients + init m/s
    gat_attn_coeff<<<cdiv((long long)n_nodes * heads, 256), 256, 0, stream>>>(
        bufH, a_src, a_dst, as, ad, mm, ss, n_nodes, heads);
    // 3) zero aggregation buffer
    gat_zero<<<cdiv((long long)n_nodes * HC, 256), 256, 0, stream>>>(bufO, (size_t)n_nodes * HC);
    // 4) segment max over incoming edges
    gat_edge_max<<<cdiv((long long)Etot * heads, 256), 256, 0, stream>>>(
        ei, E, Etot, as, ad, mm, heads);
    // 5) segment exp-sum
    gat_edge_sum<<<cdiv((long long)Etot * heads, 256), 256, 0, stream>>>(
        ei, E, Etot, as, ad, mm, ss, heads);
    // 6) weighted aggregation (wave per edge-head)
    gat_edge_aggr<<<cdiv((long long)Etot * heads * 32, 256), 256, 0, stream>>>(
        ei, E, Etot, as, ad, mm, ss, bufH, bufO, heads);
    // 7) bias + ELU
    gat_bias_elu<<<cdiv((long long)n_nodes * HC, 256), 256, 0, stream>>>(
        bufO, bias, n_nodes, HC);
}

extern "C" void kernel_launch(void* const* d_in, const int* in_sizes, int n_in,
                              void* d_out, int out_size, void* d_ws, size_t ws_size,
                              hipStream_t stream) {
    const float* x      = (const float*)d_in[0];
    const int*   ei     = (const int*)  d_in[1];
    const float* W1     = (const float*)d_in[2];
    const float* a1_src = (const float*)d_in[3];
    const float* a1_dst = (const float*)d_in[4];
    const float* b1     = (const float*)d_in[5];
    const float* W2     = (const float*)d_in[6];
    const float* a2_src = (const float*)d_in[7];
    const float* a2_dst = (const float*)d_in[8];
    const float* b2     = (const float*)d_in[9];
    const float* W3     = (const float*)d_in[10];
    const float* a3_src = (const float*)d_in[11];
    const float* a3_dst = (const float*)d_in[12];
    const float* b3     = (const float*)d_in[13];
    const float* Wc     = (const float*)d_in[14];
    const float* bc     = (const float*)d_in[15];

    const int n_nodes = in_sizes[0] / GAT_DIN;     // 50000
    const int E       = in_sizes[1] / 2;           // 800000
    const int Etot    = E + n_nodes;               // + self loops

    // Workspace layout (floats):
    //   bufH : n_nodes * 512   (GEMM output h, per layer)
    //   bufO : n_nodes * 512   (aggregated output, per layer)
    //   as, ad, mm, ss : n_nodes * 8 each
    float* ws   = (float*)d_ws;
    float* bufH = ws;
    float* bufO = bufH + (size_t)n_nodes * 512;
    float* as   = bufO + (size_t)n_nodes * 512;
    float* ad   = as + (size_t)n_nodes * GAT_HEADS;
    float* mm   = ad + (size_t)n_nodes * GAT_HEADS;
    float* ss   = mm + (size_t)n_nodes * GAT_HEADS;

    // Layer 1: 128 -> 8x64 (concat)
    run_gat_layer(x, W1, a1_src, a1_dst, b1, bufH, bufO, as, ad, mm, ss,
                  ei, E, Etot, n_nodes, GAT_DIN, GAT_HEADS, stream);
    // Layer 2: 512 -> 64 (1 head; mean over 1 head == identity)
    run_gat_layer(bufO, W2, a2_src, a2_dst, b2, bufH, bufO, as, ad, mm, ss,
                  ei, E, Etot, n_nodes, GAT_HEADS * GAT_HID, 1, stream);
    // Layer 3: 64 -> 64 (1 head)
    run_gat_layer(bufO, W3, a3_src, a3_dst, b3, bufH, bufO, as, ad, mm, ss,
                  ei, E, Etot, n_nodes, GAT_HID, 1, stream);
    // Classifier: 64 -> 10
    gat_classifier<<<cdiv((long long)n_nodes * GAT_DOUT, 256), 256, 0, stream>>>(
        bufO, Wc, bc, (float*)d_out, n_nodes);
}